// WindowAttention_31044023615515
// MI455X (gfx1250) — compile-verified
//
#include <hip/hip_runtime.h>

typedef __attribute__((ext_vector_type(2))) float v2f;
typedef __attribute__((ext_vector_type(8))) float v8f;

#define NTOK 64
#define CDIM 384
#define NH   12
#define HD   32
#define NW   64

// Row t of cat([q,k,v], axis=tokens) for batch b, columns [h*HD, h*HD+HD)
__device__ __forceinline__ const float* cat_row(const float* __restrict__ q,
                                                const float* __restrict__ k,
                                                const float* __restrict__ v,
                                                int b, int h, int t) {
    const float* base;
    int r;
    if (t < NTOK)          { base = q; r = t; }
    else if (t < 2 * NTOK) { base = k; r = t - NTOK; }
    else                   { base = v; r = t - 2 * NTOK; }
    return base + ((size_t)b * NTOK + r) * CDIM + h * HD;
}

__global__ __launch_bounds__(32)
void win_attn_fused(const float* __restrict__ q, const float* __restrict__ k,
                    const float* __restrict__ v, const float* __restrict__ bias_table,
                    const float* __restrict__ mask, const int* __restrict__ rel_index,
                    float* __restrict__ out) {
    __shared__ float Qs[NTOK * HD];   // 8 KB, pre-scaled
    __shared__ float Ks[NTOK * HD];   // 8 KB
    __shared__ float Vs[NTOK * HD];   // 8 KB
    __shared__ float Pb[16 * NTOK];   // 4 KB, one 16-row softmaxed P block

    const int h    = blockIdx.x;      // 0..11
    const int b    = blockIdx.y;      // 0..2047
    const int lane = threadIdx.x;     // wave32
    const int low  = lane & 15;
    const int hi   = lane >> 4;
    const int kb   = hi << 1;         // K sub-pair selected by lane half
    const float scale = 0.17677669529663688f; // 32^-0.5

    // ---- Stage Q (scaled), K, V into LDS with the mod-3 interleave gather ----
    for (int idx = lane; idx < 512; idx += 32) {
        int row = idx >> 3;
        int c4  = (idx & 7) << 2;
        const float4 qv = *(const float4*)(cat_row(q, k, v, b, h, 3 * row + 0) + c4);
        const float4 kv = *(const float4*)(cat_row(q, k, v, b, h, 3 * row + 1) + c4);
        const float4 vv = *(const float4*)(cat_row(q, k, v, b, h, 3 * row + 2) + c4);
        Qs[row * HD + c4 + 0] = qv.x * scale;
        Qs[row * HD + c4 + 1] = qv.y * scale;
        Qs[row * HD + c4 + 2] = qv.z * scale;
        Qs[row * HD + c4 + 3] = qv.w * scale;
        Ks[row * HD + c4 + 0] = kv.x;
        Ks[row * HD + c4 + 1] = kv.y;
        Ks[row * HD + c4 + 2] = kv.z;
        Ks[row * HD + c4 + 3] = kv.w;
        Vs[row * HD + c4 + 0] = vv.x;
        Vs[row * HD + c4 + 1] = vv.y;
        Vs[row * HD + c4 + 2] = vv.z;
        Vs[row * HD + c4 + 3] = vv.w;
    }
    __syncthreads();

    const float* mwin = mask + (size_t)(b & (NW - 1)) * NTOK * NTOK;

    for (int ti = 0; ti < 4; ++ti) {           // 16-row block of the 64x64 attention
        // ---- S = (Q*scale) @ K^T for this row block: 4 tiles of 16x16 ----
        v8f acc[4];
        #pragma unroll
        for (int tj = 0; tj < 4; ++tj) acc[tj] = (v8f)0.0f;

        #pragma unroll
        for (int kk = 0; kk < HD; kk += 4) {
            v2f a;
            a[0] = Qs[(ti * 16 + low) * HD + kk + kb];
            a[1] = Qs[(ti * 16 + low) * HD + kk + kb + 1];
            #pragma unroll
            for (int tj = 0; tj < 4; ++tj) {
                v2f bm;
                bm[0] = Ks[(tj * 16 + low) * HD + kk + kb];
                bm[1] = Ks[(tj * 16 + low) * HD + kk + kb + 1];
                acc[tj] = __builtin_amdgcn_wmma_f32_16x16x4_f32(
                    false, a, false, bm, (short)0, acc[tj], false, false);
            }
        }

        // ---- bias + mask + row softmax (rows owned per lane-half) ----
        #pragma unroll
        for (int vr = 0; vr < 8; ++vr) {
            int r = vr + 8 * hi;               // row within this 16-row block
            int n = ti * 16 + r;               // global attention row
            float s[4];
            #pragma unroll
            for (int tj = 0; tj < 4; ++tj) {
                int mcol = tj * 16 + low;
                int bidx = rel_index[n * NTOK + mcol];
                s[tj] = acc[tj][vr] + bias_table[bidx * NH + h] + mwin[n * NTOK + mcol];
            }
            float mx = fmaxf(fmaxf(s[0], s[1]), fmaxf(s[2], s[3]));
            #pragma unroll
            for (int off = 1; off < 16; off <<= 1)
                mx = fmaxf(mx, __shfl_xor(mx, off, 32));
            float p[4];
            float sum = 0.0f;
            #pragma unroll
            for (int tj = 0; tj < 4; ++tj) { p[tj] = __expf(s[tj] - mx); sum += p[tj]; }
            #pragma unroll
            for (int off = 1; off < 16; off <<= 1)
                sum += __shfl_xor(sum, off, 32);
            float inv = 1.0f / sum;
            #pragma unroll
            for (int tj = 0; tj < 4; ++tj)
                Pb[r * NTOK + tj * 16 + low] = p[tj] * inv;
        }
        __syncthreads();   // P block visible to all lanes in WMMA-A layout reads

        // ---- O_block = P_block @ V : two 16x16 output tiles ----
        #pragma unroll
        for (int toj = 0; toj < 2; ++toj) {
            v8f o = (v8f)0.0f;
            #pragma unroll
            for (int kk = 0; kk < NTOK; kk += 4) {
                v2f a;
                a[0] = Pb[low * NTOK + kk + kb];
                a[1] = Pb[low * NTOK + kk + kb + 1];
                v2f bm;
                bm[0] = Vs[(kk + kb) * HD + toj * 16 + low];
                bm[1] = Vs[(kk + kb + 1) * HD + toj * 16 + low];
                o = __builtin_amdgcn_wmma_f32_16x16x4_f32(
                    false, a, false, bm, (short)0, o, false, false);
            }
            #pragma unroll
            for (int vr = 0; vr < 8; ++vr) {
                int n = ti * 16 + vr + 8 * hi;
                out[((size_t)b * NTOK + n) * CDIM + h * HD + toj * 16 + low] = o[vr];
            }
        }
        __syncthreads();   // done reading Pb before next ti overwrites it
    }
}

extern "C" void kernel_launch(void* const* d_in, const int* in_sizes, int n_in,
                              void* d_out, int out_size, void* d_ws, size_t ws_size,
                              hipStream_t stream) {
    (void)in_sizes; (void)n_in; (void)d_ws; (void)ws_size; (void)out_size;
    const float* q          = (const float*)d_in[0];
    const float* k          = (const float*)d_in[1];
    const float* v          = (const float*)d_in[2];
    const float* bias_table = (const float*)d_in[3];
    const float* mask       = (const float*)d_in[4];
    const int*   rel_index  = (const int*)d_in[5];
    float* out = (float*)d_out;

    dim3 grid(NH, 2048, 1);   // one wave32 per (head, window-batch)
    dim3 block(32, 1, 1);
    win_attn_fused<<<grid, block, 0, stream>>>(q, k, v, bias_table, mask, rel_index, out);
}